// QLSTM_Enhanced_65481071409492
// MI455X (gfx1250) — compile-verified
//
#include <hip/hip_runtime.h>
#include <math.h>

// ---------------------------------------------------------------------------
// LSTM (T=1024, B=64, D_IN=256, D_H=512) for MI455X (gfx1250, wave32).
// Persistent-kernel design: 32 WGs, each owns 16 h-columns (= 64 gate-aligned
// z columns). Per step: stage [x_t | h_prev] in LDS, fp32 WMMA 16x16x4 GEMM,
// native-trans gates, c-state in registers, h double-buffered in d_ws,
// per-step grid barrier. Weights pre-packed into per-lane WMMA fragment order
// (one global_load_b128 per K-iter) when workspace allows; direct-load
// fallback otherwise.
// ---------------------------------------------------------------------------

#define T_STEPS 1024
#define BATCH   64
#define DIN     256
#define DH      512
#define KTOT    (DIN + DH)      // 768
#define N4H     (4 * DH)        // 2048
#define NWG     32              // workgroups (one per WGP)
#define HC      16              // h-columns per WG
#define THREADS 256             // 8 wave32s
#define KITERS  (KTOT / 4)      // 192

#define COMB_STRIDE 772         // 768 + 4 floats pad -> conflict-free column reads
#define ZB_STRIDE   69          // 64 + 5 floats pad

typedef __attribute__((ext_vector_type(2))) float v2f;
typedef __attribute__((ext_vector_type(8))) float v8f;

// workspace layout
#define WS_HBUF_BYTES   ((size_t)2 * BATCH * DH * sizeof(float))   // 262144
#define WS_CNT_BYTES    ((size_t)4096)                              // T ints, padded
#define WS_WPACK_BYTES  ((size_t)KTOT * N4H * sizeof(float))        // 6291456
#define WS_NEED_PACKED  (WS_HBUF_BYTES + WS_CNT_BYTES + WS_WPACK_BYTES)

// Branch-free fast sigmoid / tanh on native v_exp_f32.
__device__ __forceinline__ float fast_sigmoid(float x) {
  return 1.0f / (1.0f + __expf(-x));
}
__device__ __forceinline__ float fast_tanh(float x) {
  float e = __expf(-2.0f * fabsf(x));       // in (0,1], no overflow
  float r = (1.0f - e) / (1.0f + e);
  return copysignf(r, x);
}

__global__ void lstm_init_kernel(float* __restrict__ hbuf0,
                                 int* __restrict__ counters) {
  int idx = blockIdx.x * blockDim.x + threadIdx.x;
  int stride = gridDim.x * blockDim.x;
  for (int i = idx; i < BATCH * DH; i += stride) hbuf0[i] = 0.0f;
  for (int i = idx; i < T_STEPS; i += stride) counters[i] = 0;
}

// Repack W4 [768][2048] row-major into per-lane WMMA fragment order:
//   Wp[(((wg*2 + pair)*KITERS + kk)*32 + lane)] = float4{
//       W4[kb][c0], W4[kb+1][c0], W4[kb][c0+512], W4[kb+1][c0+512] }
//   kb = 4*kk + 2*(lane>>4),  c0 = (2*pair)*512 + wg*16 + (lane&15)
__global__ void lstm_pack_w_kernel(const float* __restrict__ W4,
                                   float4* __restrict__ Wp) {
  const int total = NWG * 2 * KITERS * 32;            // 393216 float4s
  int stride = gridDim.x * blockDim.x;
  for (int idx = blockIdx.x * blockDim.x + threadIdx.x; idx < total;
       idx += stride) {
    int lane = idx & 31;
    int q    = idx >> 5;
    int kk   = q % KITERS;
    int pair = (q / KITERS) & 1;
    int wg   = q / (KITERS * 2);
    int kb   = kk * 4 + ((lane >> 4) << 1);
    int c0   = (pair * 2) * DH + wg * HC + (lane & 15);
    float4 v;
    v.x = W4[(size_t)kb * N4H + c0];
    v.y = W4[(size_t)(kb + 1) * N4H + c0];
    v.z = W4[(size_t)kb * N4H + c0 + DH];
    v.w = W4[(size_t)(kb + 1) * N4H + c0 + DH];
    Wp[idx] = v;
  }
}

template <bool PACKED>
__global__ __launch_bounds__(THREADS, 1)
void lstm_persistent_kernel(const float* __restrict__ x,      // [T][B][DIN]
                            const float* __restrict__ W4,     // [KTOT][4*DH]
                            const float4* __restrict__ Wpack, // packed fragments
                            const float* __restrict__ b4,     // [4*DH]
                            float* __restrict__ out,          // stacked | hx | cx
                            float* __restrict__ hbuf,         // [2][B*DH] in ws
                            int* __restrict__ counters) {     // [T] in ws
  __shared__ float comb[BATCH * COMB_STRIDE];   // [64][772] : [x_t | h_prev]
  __shared__ float zbuf[BATCH * ZB_STRIDE];     // [64][69]  : z slice

  const int tid   = threadIdx.x;
  const int wg    = blockIdx.x;           // 0..31
  const int lane  = tid & 31;
  const int wave  = tid >> 5;
  const int hcol0 = wg * HC;

  // --- WMMA tile assignment: wave -> 1 M-tile x 2 N-tiles (two gates) ------
  const int mt  = wave & 3;               // M tile
  const int ntp = (wave >> 2) * 2;        // first N tile == first gate index
  const int l15 = lane & 15;
  const int ak  = (lane >> 4) * 2;        // K sub-offset (ISA lane-half split)
  const int am  = mt * 16 + l15;          // A row (batch index) for this lane
  const float*  pW  = W4 + (size_t)(ntp * DH + hcol0 + l15);
  const float4* pWp = Wpack +
      ((size_t)(wg * 2 + (wave >> 2)) * KITERS) * 32 + lane;

  // --- elementwise assignment: thread -> (m = tid&63, 4 consecutive cols) --
  const int em  = tid & 63;
  const int ec0 = (tid >> 6) * 4;
  float creg[4] = {0.0f, 0.0f, 0.0f, 0.0f};
  float bf[4], bi[4], bg[4], bo[4];
  for (int j = 0; j < 4; ++j) {
    int col = hcol0 + ec0 + j;
    bf[j] = b4[0 * DH + col];
    bi[j] = b4[1 * DH + col];
    bg[j] = b4[2 * DH + col];
    bo[j] = b4[3 * DH + col];
  }

  for (int t = 0; t < T_STEPS; ++t) {
    const float* hprev = hbuf + (size_t)(t & 1) * (BATCH * DH);
    float*       hnext = hbuf + (size_t)((t + 1) & 1) * (BATCH * DH);

    // ---- stage combined = [x_t | h_prev] into LDS (coalesced float4) ------
    const float4* x4 = (const float4*)(x + (size_t)t * BATCH * DIN);
    for (int i = tid; i < BATCH * (DIN / 4); i += THREADS) {
      int m = i >> 6, kq = i & 63;
      *(float4*)&comb[m * COMB_STRIDE + kq * 4] = x4[m * (DIN / 4) + kq];
    }
    const float4* h4 = (const float4*)hprev;
    for (int i = tid; i < BATCH * (DH / 4); i += THREADS) {
      int m = i >> 7, kq = i & 127;
      *(float4*)&comb[m * COMB_STRIDE + DIN + kq * 4] = h4[m * (DH / 4) + kq];
    }
    __syncthreads();

    // ---- z = combined @ W4 via V_WMMA_F32_16X16X4_F32 ---------------------
    v8f acc0 = {0.f, 0.f, 0.f, 0.f, 0.f, 0.f, 0.f, 0.f};
    v8f acc1 = acc0;
    if (PACKED) {
      const float4* wp = pWp;
      for (int k = 0; k < KTOT; k += 4) {
        v2f a = *(const v2f*)&comb[am * COMB_STRIDE + k + ak];
        float4 bq = *wp;            // one coalesced b128 per K-iter
        wp += 32;
        v2f b0 = {bq.x, bq.y};
        v2f b1 = {bq.z, bq.w};
        acc0 = __builtin_amdgcn_wmma_f32_16x16x4_f32(
                   false, a, false, b0, (short)0, acc0, false, false);
        acc1 = __builtin_amdgcn_wmma_f32_16x16x4_f32(
                   false, a, false, b1, (short)0, acc1, false, false);
      }
    } else {
      for (int k = 0; k < KTOT; k += 4) {
        const int kb = k + ak;
        v2f a = *(const v2f*)&comb[am * COMB_STRIDE + kb];
        const float* w = pW + (size_t)kb * N4H;
        v2f b0 = { w[0],  w[N4H] };
        v2f b1 = { w[DH], w[N4H + DH] };
        acc0 = __builtin_amdgcn_wmma_f32_16x16x4_f32(
                   false, a, false, b0, (short)0, acc0, false, false);
        acc1 = __builtin_amdgcn_wmma_f32_16x16x4_f32(
                   false, a, false, b1, (short)0, acc1, false, false);
      }
    }

    // ---- scatter accumulators to LDS (C/D: N=lane&15, M=r+8*(lane>=16)) ---
    {
      int mrow = mt * 16 + ((lane >> 4) << 3);
      int n0   = ntp * 16 + l15;
      for (int r = 0; r < 8; ++r) {
        zbuf[(mrow + r) * ZB_STRIDE + n0]      = acc0[r];
        zbuf[(mrow + r) * ZB_STRIDE + n0 + 16] = acc1[r];
      }
    }
    __syncthreads();

    // ---- gates + state update (native v_exp_f32, c in registers) ----------
    for (int j = 0; j < 4; ++j) {
      int c = ec0 + j;
      float zf = zbuf[em * ZB_STRIDE + c +  0] + bf[j];
      float zi = zbuf[em * ZB_STRIDE + c + 16] + bi[j];
      float zg = zbuf[em * ZB_STRIDE + c + 32] + bg[j];
      float zo = zbuf[em * ZB_STRIDE + c + 48] + bo[j];
      float fg = fast_sigmoid(zf);
      float ig = fast_sigmoid(zi);
      float gg = fast_tanh(zg);
      float og = fast_sigmoid(zo);
      float cn = fg * creg[j] + ig * gg;
      float hn = og * fast_tanh(cn);
      creg[j]  = cn;
      int col  = hcol0 + c;
      hnext[em * DH + col] = hn;
      out[(size_t)t * (BATCH * DH) + em * DH + col] = hn;
      if (t == T_STEPS - 1) {
        out[(size_t)T_STEPS * BATCH * DH + em * DH + col]              = hn;
        out[(size_t)T_STEPS * BATCH * DH + BATCH * DH + em * DH + col] = cn;
      }
    }

    // ---- grid-wide barrier: fresh counter slot per step -------------------
    __threadfence();
    __syncthreads();
    if (tid == 0) {
      __hip_atomic_fetch_add(&counters[t], 1, __ATOMIC_ACQ_REL,
                             __HIP_MEMORY_SCOPE_AGENT);
      while (__hip_atomic_load(&counters[t], __ATOMIC_ACQUIRE,
                               __HIP_MEMORY_SCOPE_AGENT) < NWG) {
        __builtin_amdgcn_s_sleep(1);
      }
    }
    __syncthreads();
    __threadfence();
  }
}

extern "C" void kernel_launch(void* const* d_in, const int* in_sizes, int n_in,
                              void* d_out, int out_size, void* d_ws, size_t ws_size,
                              hipStream_t stream) {
  const float* x  = (const float*)d_in[0];   // [T][B][DIN]
  const float* W4 = (const float*)d_in[1];   // [KTOT][4*DH]
  const float* b4 = (const float*)d_in[2];   // [4*DH]
  float* out = (float*)d_out;

  float*  hbuf     = (float*)d_ws;
  int*    counters = (int*)((char*)d_ws + WS_HBUF_BYTES);
  float4* Wp       = (float4*)((char*)d_ws + WS_HBUF_BYTES + WS_CNT_BYTES);

  lstm_init_kernel<<<8, 256, 0, stream>>>(hbuf, counters);

  if (ws_size >= WS_NEED_PACKED) {
    lstm_pack_w_kernel<<<256, 256, 0, stream>>>(W4, Wp);
    lstm_persistent_kernel<true><<<NWG, THREADS, 0, stream>>>(
        x, W4, Wp, b4, out, hbuf, counters);
  } else {
    lstm_persistent_kernel<false><<<NWG, THREADS, 0, stream>>>(
        x, W4, Wp, b4, out, hbuf, counters);
  }
}